// SinkhornLoss_9629316677892
// MI455X (gfx1250) — compile-verified
//
#include <hip/hip_runtime.h>

typedef float v2f __attribute__((ext_vector_type(2)));
typedef float v8f __attribute__((ext_vector_type(8)));

#define NB 32
#define NDIM 1024
#define WAVES 8
#define BLOCK (WAVES * 32)
#define TILE_O 16
#define TILE_R 64
#define LDS_PITCH 68                      // 64 + 4 pad -> conflict-free ds_load_b64
#define WAVE_LDS (TILE_O * LDS_PITCH + TILE_R)
#define NEG10LOG2E (-14.426950408889634f) // exp(-D/0.1) == exp2(NEG10LOG2E * D)

// raw v_exp_f32: input range [-14.43, 0] needs no denorm fixup
__device__ __forceinline__ float k0_of(float d) {
    return __builtin_amdgcn_exp2f(NEG10LOG2E * d);
}

__global__ __launch_bounds__(256) void sink_init(float* bvec, float* out) {
    int idx = blockIdx.x * blockDim.x + threadIdx.x;
    bvec[idx] = 1.0f;
    if (idx == 0) out[0] = 0.0f;
}

// out[o] = 1 / sum_r K0(o,r) * vin[r]
// ROW pass (COLPASS=false): o = row, r = col  -> a = 1/(K0 b)
// COL pass (COLPASS=true):  o = col, r = row  -> b = 1/(K0^T a)
template <bool COLPASS>
__global__ __launch_bounds__(BLOCK) void sink_pass(const float* __restrict__ D,
                                                   const float* __restrict__ vin,
                                                   float* __restrict__ vout) {
    __shared__ float smem[WAVES * WAVE_LDS];

    const int wave = threadIdx.x >> 5;
    const int lane = threadIdx.x & 31;
    const int b    = blockIdx.x >> 3;                                // 8 blocks / batch
    const int oblk = (blockIdx.x & 7) * (WAVES * TILE_O) + wave * TILE_O;

    const float* Db = D + (size_t)b * NDIM * NDIM;
    const float* vb = vin + b * NDIM;

    float* tile = smem + wave * WAVE_LDS;          // [16][LDS_PITCH] K0 tile
    float* vlds = tile + TILE_O * LDS_PITCH;       // [64] vector chunk

    const int hi16 = lane >> 4;        // 0: lanes 0-15, 1: lanes 16-31
    const int lsub = lane & 15;
    const int kdel = hi16 * 2;         // K offset per WMMA f32 A/B lane layout

    // wave-relative LDS byte offset of this lane's 8B slot in the vector chunk
    const unsigned vchunk_lds = (unsigned)(uintptr_t)vlds + (unsigned)lane * 8u;

    v8f acc = {};                      // 16x16 f32 accumulator (8 VGPRs)

    for (int rc = 0; rc < NDIM; rc += TILE_R) {
        // ---- async DMA the 64-float vector chunk straight into LDS ----
        {
            const float* gsrc = vb + rc + lane * 2;
            asm volatile("global_load_async_to_lds_b64 %0, %1, off"
                         :: "v"(vchunk_lds), "v"(gsrc) : "memory");
        }

        // ---- stage K0 tile (16 x 64, exp applied) via b128 loads ----
        if (!COLPASS) {
            #pragma unroll
            for (int t = 0; t < 8; ++t) {
                int idx4 = t * 32 + lane;           // 0..255 float4 slots
                int o = idx4 >> 4;                  // 16 float4 per 64-float row
                int r = (idx4 & 15) << 2;
                size_t ga = (size_t)(oblk + o) * NDIM + (size_t)(rc + r);
                float4 d4 = *(const float4*)(Db + ga);
                if (rc + TILE_R < NDIM) __builtin_prefetch(Db + ga + TILE_R, 0, 1);
                float4 k4 = make_float4(k0_of(d4.x), k0_of(d4.y), k0_of(d4.z), k0_of(d4.w));
                *(float4*)(tile + o * LDS_PITCH + r) = k4;
            }
        } else {
            #pragma unroll
            for (int t = 0; t < 8; ++t) {
                int idx4 = t * 32 + lane;
                int r  = idx4 >> 2;                 // 4 float4 per 16-col row
                int o4 = (idx4 & 3) << 2;
                size_t ga = (size_t)(rc + r) * NDIM + (size_t)(oblk + o4);
                float4 d4 = *(const float4*)(Db + ga);
                if (rc + TILE_R < NDIM)
                    __builtin_prefetch(Db + ga + (size_t)TILE_R * NDIM, 0, 1);
                tile[(o4 + 0) * LDS_PITCH + r] = k0_of(d4.x);   // transposed scatter,
                tile[(o4 + 1) * LDS_PITCH + r] = k0_of(d4.y);   // conflict-free banks
                tile[(o4 + 2) * LDS_PITCH + r] = k0_of(d4.z);
                tile[(o4 + 3) * LDS_PITCH + r] = k0_of(d4.w);
            }
        }

        asm volatile("s_wait_asynccnt 0x0" ::: "memory");
        __syncthreads();

        // ---- 16 WMMAs: D += A(16x4) x B(4x16), K marches along the 64-chunk ----
        #pragma unroll
        for (int k0 = 0; k0 < TILE_R; k0 += 4) {
            v2f tOp = *(const v2f*)(tile + lsub * LDS_PITCH + k0 + kdel); // K0 tile
            v2f wOp = *(const v2f*)(vlds + k0 + kdel);                    // broadcast vec
            if (COLPASS)
                acc = __builtin_amdgcn_wmma_f32_16x16x4_f32(
                        false, wOp, false, tOp, (short)0, acc, false, false);
            else
                acc = __builtin_amdgcn_wmma_f32_16x16x4_f32(
                        false, tOp, false, wOp, (short)0, acc, false, false);
        }
        __syncthreads();
    }

    // ---- extract duplicated reduction result, write reciprocal dual ----
    float* ob = vout + b * NDIM + oblk;
    if (COLPASS) {
        // D[m][n] identical over m; VGPR0 lanes 0-15 hold n = 0..15
        if (lane < 16) ob[lane] = 1.0f / acc[0];
    } else {
        // D[m][n] identical over n; lane 0 has rows 0-7, lane 16 rows 8-15
        if (lsub == 0) {
            #pragma unroll
            for (int v = 0; v < 8; ++v) ob[hi16 * 8 + v] = 1.0f / acc[v];
        }
    }
}

__global__ __launch_bounds__(256) void sink_loss(const float* __restrict__ D,
                                                 const float* __restrict__ avec,
                                                 const float* __restrict__ bvec,
                                                 float* __restrict__ out) {
    __shared__ float red[256];
    const int b       = blockIdx.x >> 3;
    const int rowbase = (blockIdx.x & 7) * 128;
    const float* Db = D + (size_t)b * NDIM * NDIM + (size_t)rowbase * NDIM;
    const float* av = avec + b * NDIM + rowbase;
    const float* bv = bvec + b * NDIM;

    float s = 0.0f;
    for (int idx4 = threadIdx.x; idx4 < (128 * NDIM) / 4; idx4 += 256) {
        int i  = idx4 >> 8;            // 256 float4 per 1024-col row
        int j4 = (idx4 & 255) << 2;
        float4 d4 = *(const float4*)(Db + (size_t)idx4 * 4);
        float ai = av[i];
        s += k0_of(d4.x) * d4.x * ai * bv[j4 + 0];
        s += k0_of(d4.y) * d4.y * ai * bv[j4 + 1];
        s += k0_of(d4.z) * d4.z * ai * bv[j4 + 2];
        s += k0_of(d4.w) * d4.w * ai * bv[j4 + 3];
    }
    red[threadIdx.x] = s;
    __syncthreads();
    for (int off = 128; off; off >>= 1) {
        if (threadIdx.x < off) red[threadIdx.x] += red[threadIdx.x + off];
        __syncthreads();
    }
    if (threadIdx.x == 0) atomicAdd(out, red[0] * (1.0f / NB));
}

extern "C" void kernel_launch(void* const* d_in, const int* in_sizes, int n_in,
                              void* d_out, int out_size, void* d_ws, size_t ws_size,
                              hipStream_t stream) {
    const float* D = (const float*)d_in[0];
    float* out  = (float*)d_out;
    float* avec = (float*)d_ws;             // 32*1024 f32
    float* bvec = avec + NB * NDIM;         // 32*1024 f32 (total 256 KB of d_ws)

    sink_init<<<(NB * NDIM) / 256, 256, 0, stream>>>(bvec, out);
    for (int it = 0; it < 20; ++it) {
        sink_pass<false><<<NB * 8, BLOCK, 0, stream>>>(D, bvec, avec); // a = 1/(K0 b)
        sink_pass<true ><<<NB * 8, BLOCK, 0, stream>>>(D, avec, bvec); // b = 1/(K0^T a)
    }
    sink_loss<<<NB * 8, 256, 0, stream>>>(D, avec, bvec, out);
}